// SRUCell_74783970558448
// MI455X (gfx1250) — compile-verified
//
#include <hip/hip_runtime.h>

typedef __attribute__((ext_vector_type(2))) float v2f;
typedef __attribute__((ext_vector_type(8))) float v8f;

#define SRU_L      1024
#define SRU_B      16
#define SRU_D      2048
#define SRU_HID    2048
#define SRU_NC     (3 * SRU_HID)   // 6144 columns of U / weight

// Async global->LDS 16-byte copy (gfx1250, ASYNCcnt-tracked).
__device__ __forceinline__ void async_copy_b128(unsigned ldsaddr, const void* gaddr) {
    asm volatile("global_load_async_to_lds_b128 %0, %1, off"
                 :: "v"(ldsaddr), "v"(gaddr) : "memory");
}
__device__ __forceinline__ void wait_async0() {
    asm volatile("s_wait_asynccnt 0x0" ::: "memory");
}

// ---------------------------------------------------------------------------
// GEMM:  U[rows x 6144] = A[rows x 2048] * W[2048 x 6144]   (all row-major)
// 256 threads = 8 waves, block tile 64(M) x 128(N); waves in 2(M) x 4(N) grid,
// each wave owns a 32x32 register tile (4x v8f accumulators) -> 4 WMMA per
// k-step against 2 A + 2 B fragment loads. K staged 32-wide through LDS with
// DOUBLE-BUFFERED async global->LDS b128 copies (copy slab k+1 || compute k).
// ---------------------------------------------------------------------------
__global__ __launch_bounds__(256) void sru_gemm(const float* __restrict__ A,
                                                const float* __restrict__ W,
                                                float* __restrict__ U) {
    const int K = SRU_D;
    const int N = SRU_NC;

    __shared__ float As[2][64][36];    // row stride 144B: 16B-aligned, bank-spread
    __shared__ float Bs[2][32][132];   // row stride 528B: 16B-aligned, bank-split halves

    const int t    = threadIdx.x;      // 0..255
    const int wave = t >> 5;           // 0..7
    const int lane = t & 31;
    const int wm   = wave >> 2;        // 0..1  -> M sub-block (32 rows)
    const int wn   = wave & 3;         // 0..3  -> N sub-block (32 cols)

    const int rowBase = blockIdx.y * 64;
    const int colBase = blockIdx.x * 128;

    // A staging: 64 x 32 = 2048 floats; 8 per thread (2x b128)
    const int a_r = t >> 2;            // 0..63
    const int a_c = (t & 3) << 3;      // 0,8,16,24
    // B staging: 32 x 128 = 4096 floats; 16 per thread (4x b128)
    const int b_c = (t & 31) << 2;     // 0,4,..,124
    const int b_r = t >> 5;            // 0..7 (+ {0,8,16,24})

    // WMMA fragment indexing (ISA 7.12.2, wave32)
    const int ml      = lane & 15;
    const int halfsel = (lane >> 4) << 1;     // K offset {0,2} for v0/v1
    const int mbase   = (lane >> 4) << 3;     // C/D: M = vreg + {0,8}

    const float* Arow = A + (size_t)(rowBase + a_r) * K + a_c;
    const float* Brow = W + colBase + b_c;

    unsigned asA0[2], asA1[2], asB[2][4];
#pragma unroll
    for (int p = 0; p < 2; ++p) {
        asA0[p] = (unsigned)(size_t)&As[p][a_r][a_c];
        asA1[p] = (unsigned)(size_t)&As[p][a_r][a_c + 4];
#pragma unroll
        for (int i = 0; i < 4; ++i)
            asB[p][i] = (unsigned)(size_t)&Bs[p][b_r + (i << 3)][b_c];
    }

    const int m0 = wm << 5;
    const int n0 = wn << 5;

    v8f acc00 = {}, acc01 = {}, acc10 = {}, acc11 = {};

    // ---- prologue: stage slab 0 into buffer 0 ----
    async_copy_b128(asA0[0], Arow);
    async_copy_b128(asA1[0], Arow + 4);
#pragma unroll
    for (int i = 0; i < 4; ++i)
        async_copy_b128(asB[0][i], Brow + (size_t)(b_r + (i << 3)) * N);
    wait_async0();
    __syncthreads();

    int p = 0;
    for (int k0 = 0; k0 < K; k0 += 32) {
        // ---- issue async copies for slab k0+32 into the other buffer ----
        const int q = p ^ 1;
        if (k0 + 32 < K) {
            const int kn = k0 + 32;
            async_copy_b128(asA0[q], Arow + kn);
            async_copy_b128(asA1[q], Arow + kn + 4);
#pragma unroll
            for (int i = 0; i < 4; ++i)
                async_copy_b128(asB[q][i], Brow + (size_t)(kn + b_r + (i << 3)) * N);
        }

        // ---- compute on buffer p: 8 k-steps x 4 WMMA ----
#pragma unroll
        for (int kk = 0; kk < 32; kk += 4) {
            v2f a0, a1, b0, b1;
            a0.x = As[p][m0 + ml][kk + halfsel];
            a0.y = As[p][m0 + ml][kk + halfsel + 1];
            a1.x = As[p][m0 + 16 + ml][kk + halfsel];
            a1.y = As[p][m0 + 16 + ml][kk + halfsel + 1];
            b0.x = Bs[p][kk + halfsel][n0 + ml];
            b0.y = Bs[p][kk + halfsel + 1][n0 + ml];
            b1.x = Bs[p][kk + halfsel][n0 + 16 + ml];
            b1.y = Bs[p][kk + halfsel + 1][n0 + 16 + ml];
            acc00 = __builtin_amdgcn_wmma_f32_16x16x4_f32(false, a0, false, b0, (short)0, acc00, false, false);
            acc01 = __builtin_amdgcn_wmma_f32_16x16x4_f32(false, a0, false, b1, (short)0, acc01, false, false);
            acc10 = __builtin_amdgcn_wmma_f32_16x16x4_f32(false, a1, false, b0, (short)0, acc10, false, false);
            acc11 = __builtin_amdgcn_wmma_f32_16x16x4_f32(false, a1, false, b1, (short)0, acc11, false, false);
        }

        // all waves done reading buf p; this wave's copies into buf q landed
        wait_async0();
        __syncthreads();
        p = q;
    }

    // ---- write the four 16x16 tiles ----
    float* C00 = U + (size_t)(rowBase + m0 + mbase) * N + colBase + n0 + ml;
#pragma unroll
    for (int r = 0; r < 8; ++r) {
        C00[(size_t)r * N]             = acc00[r];
        C00[(size_t)r * N + 16]        = acc01[r];
        C00[(size_t)(r + 16) * N]      = acc10[r];
        C00[(size_t)(r + 16) * N + 16] = acc11[r];
    }
}

// ---------------------------------------------------------------------------
// Sequential scan over curL timesteps. One thread per (b, j); c carried in a
// register; (u0,u1,u2) are contiguous 12-byte triplets (gate-interleaved).
// ---------------------------------------------------------------------------
__global__ __launch_bounds__(256) void sru_scan(const float* __restrict__ U,
                                                const float* __restrict__ x,
                                                const float* __restrict__ wc,
                                                const float* __restrict__ bias,
                                                float* __restrict__ h,
                                                float* __restrict__ state,
                                                float* __restrict__ c_last,
                                                int curL, int isFirst, int isLast) {
    const int tid = blockIdx.x * blockDim.x + threadIdx.x;  // 0..32767
    const int b = tid >> 11;
    const int j = tid & (SRU_HID - 1);

    const float wcf = wc[j];
    const float wcr = wc[SRU_HID + j];
    const float bf  = bias[j];
    const float br  = bias[SRU_HID + j];
    const float scale_x = 1.7320508075688772f;  // sqrt(1 + 2*exp(HIGHWAY_BIAS)), bias = 0

    float c = isFirst ? 0.0f : state[tid];

    for (int l = 0; l < curL; ++l) {
        const size_t rb = (size_t)l * SRU_B + b;
        const float* u = U + rb * SRU_NC + 3 * j;
        const float u0 = u[0];
        const float u1 = u[1] + bf;
        const float u2 = u[2] + br;
        const float xv = x[rb * SRU_D + j] * scale_x;

        const float f = 1.0f / (1.0f + __expf(-(u1 + c * wcf)));
        c = (c - u0) * f + u0;
        const float r = 1.0f / (1.0f + __expf(-(u2 + c * wcr)));
        h[rb * SRU_HID + j] = (c - xv) * r + xv;
    }

    state[tid] = c;
    if (isLast) c_last[tid] = c;
}

// ---------------------------------------------------------------------------
// Launch: chunk over L based on ws_size; carry state at front of d_ws, U after.
// chunkL kept a multiple of 4 so GEMM rows are a multiple of 64.
// ---------------------------------------------------------------------------
extern "C" void kernel_launch(void* const* d_in, const int* in_sizes, int n_in,
                              void* d_out, int out_size, void* d_ws, size_t ws_size,
                              hipStream_t stream) {
    (void)in_sizes; (void)n_in; (void)out_size;

    const float* x      = (const float*)d_in[0];  // (L, B, D)
    const float* weight = (const float*)d_in[1];  // (D, 3*HID)
    const float* wc     = (const float*)d_in[2];  // (2*HID)
    const float* bias   = (const float*)d_in[3];  // (2*HID)

    float* out    = (float*)d_out;
    float* h      = out;                                          // L*B*HID
    float* c_last = out + (size_t)SRU_L * SRU_B * SRU_HID;        // B*HID

    const size_t stateBytes = (size_t)SRU_B * SRU_HID * sizeof(float);  // 128 KB
    float* state = (float*)d_ws;
    float* U     = state + (size_t)SRU_B * SRU_HID;

    const size_t bytesPerStep = (size_t)SRU_B * SRU_NC * sizeof(float); // 384 KB / timestep
    size_t availU = (ws_size > stateBytes) ? (ws_size - stateBytes) : 0;
    long maxChunk = (long)(availU / bytesPerStep);
    int chunkL = (maxChunk >= SRU_L) ? SRU_L : (int)(maxChunk & ~3L);
    if (chunkL < 4) chunkL = 4;

    for (int l0 = 0; l0 < SRU_L; l0 += chunkL) {
        const int curL = (SRU_L - l0 < chunkL) ? (SRU_L - l0) : chunkL;
        const int rows = curL * SRU_B;             // multiple of 64

        dim3 gGemm(SRU_NC / 128, rows / 64);
        sru_gemm<<<gGemm, 256, 0, stream>>>(x + (size_t)l0 * SRU_B * SRU_D, weight, U);

        const int isFirst = (l0 == 0);
        const int isLast  = (l0 + curL >= SRU_L);
        sru_scan<<<(SRU_B * SRU_HID) / 256, 256, 0, stream>>>(
            U, x + (size_t)l0 * SRU_B * SRU_D, wc, bias,
            h + (size_t)l0 * SRU_B * SRU_HID, state, c_last,
            curL, isFirst, isLast);
    }
}